// ChamferDist_14027363188883
// MI455X (gfx1250) — compile-verified
//
#include <hip/hip_runtime.h>

typedef float v2f __attribute__((ext_vector_type(2)));
typedef float v8f __attribute__((ext_vector_type(8)));

#define WAVES_PER_BLOCK 8
#define ROWS_PER_WAVE   16
#define ROWS_PER_BLOCK  (WAVES_PER_BLOCK * ROWS_PER_WAVE)  // 128
#define BLOCK_THREADS   (WAVES_PER_BLOCK * 32)             // 256
#define REF_CHUNK       2048                               // pts staged in LDS per pass (32 KB)

// For each query point (row), find min squared L2 distance + argmin over all ref points.
// d(q,r) = |q|^2 + |r|^2 - 2 q.r. WMMA computes  -2 q.r + |r|^2  per 16x16 tile via K=4:
//   A row  = (qx, qy, qz, 1)
//   B col  = (-2rx, -2ry, -2rz, |r|^2)
// |q|^2 is a per-row constant -> added at writeback (doesn't affect argmin).
__global__ __launch_bounds__(BLOCK_THREADS)
void chamfer_nn_wmma(const float* __restrict__ query,  // (B, nq, 3)
                     const float* __restrict__ ref,    // (B, nr, 3)
                     int nq, int nr,
                     float* __restrict__ dist_out,     // (B, nq)
                     int*   __restrict__ idx_out)      // (B, nq)
{
    __shared__ __align__(16) float ref4[REF_CHUNK * 4];   // (-2x,-2y,-2z,|r|^2) per point
    __shared__ float qsq[ROWS_PER_BLOCK];                 // |q|^2 per block row

    const int b    = blockIdx.y;
    const int tid  = threadIdx.x;
    const int lane = tid & 31;
    const int wave = tid >> 5;
    const int half = lane >> 4;   // 0: lanes 0-15 (K0,K1), 1: lanes 16-31 (K2,K3)
    const int l16  = lane & 15;

    const float* qb = query + (size_t)b * nq * 3;
    const float* rb = ref   + (size_t)b * nr * 3;

    // Per-row |q|^2 for this block (128 rows).
    const int row0 = blockIdx.x * ROWS_PER_BLOCK;
    if (tid < ROWS_PER_BLOCK) {
        const int n = row0 + tid;
        const float x = qb[n * 3 + 0], y = qb[n * 3 + 1], z = qb[n * 3 + 2];
        qsq[tid] = x * x + y * y + z * z;
    }

    // A fragment (16x4 f32): lane L%16 = row M; VGPR slot v + 2*half = K.
    const int qrow = row0 + wave * ROWS_PER_WAVE + l16;
    const float qx = qb[qrow * 3 + 0];
    const float qy = qb[qrow * 3 + 1];
    const float qz = qb[qrow * 3 + 2];
    v2f afrag;
    afrag.x = half ? qz   : qx;   // K2 : K0
    afrag.y = half ? 1.0f : qy;   // K3 : K1

    // Running (min, argmin) per C slot: slot r at lane L = element (M=r+8*half, N=l16).
    float minv[8];
    int   mini[8];
#pragma unroll
    for (int r = 0; r < 8; ++r) { minv[r] = __FLT_MAX__; mini[r] = 0; }

    for (int base = 0; base < nr; base += REF_CHUNK) {
        __syncthreads();
        // Stage preprocessed chunk of ref points into LDS.
        for (int j = tid; j < REF_CHUNK; j += BLOCK_THREADS) {
            const int g = base + j;
            const float x = rb[g * 3 + 0], y = rb[g * 3 + 1], z = rb[g * 3 + 2];
            ((float4*)ref4)[j] = make_float4(-2.0f * x, -2.0f * y, -2.0f * z,
                                             x * x + y * y + z * z);
        }
        __syncthreads();

        for (int t = 0; t < REF_CHUNK; t += 16) {
            const int jl = t + l16;                 // this lane's ref point (column N)
            // B fragment (4x16 f32): lane = N; VGPR slot v + 2*half = K.
            // ds_load_b64: lanes 0-15 -> (-2x,-2y); lanes 16-31 -> (-2z,|r|^2).
            const v2f bfrag = *(const v2f*)(ref4 + jl * 4 + half * 2);

            v8f c = {};
            c = __builtin_amdgcn_wmma_f32_16x16x4_f32(
                    false, afrag, false, bfrag, (short)0, c, false, false);

            const int gidx = base + jl;
#pragma unroll
            for (int r = 0; r < 8; ++r) {
                const float v  = c[r];
                const bool  lt = v < minv[r];       // strict: keeps earliest tile on ties
                minv[r] = lt ? v    : minv[r];
                mini[r] = lt ? gidx : mini[r];
            }
        }
    }

    // Reduce over the 16 columns within each lane-half (rows stay put).
#pragma unroll
    for (int off = 1; off < 16; off <<= 1) {
#pragma unroll
        for (int r = 0; r < 8; ++r) {
            const float ov = __shfl_xor(minv[r], off, 32);
            const int   oi = __shfl_xor(mini[r], off, 32);
            const bool take = (ov < minv[r]) || (ov == minv[r] && oi < mini[r]);
            minv[r] = take ? ov : minv[r];
            mini[r] = take ? oi : mini[r];
        }
    }

    // Writers: lane 0 -> rows M=0..7, lane 16 -> rows M=8..15 of this wave's tile.
    if (l16 == 0) {
#pragma unroll
        for (int r = 0; r < 8; ++r) {
            const int lrow = wave * ROWS_PER_WAVE + half * 8 + r;
            const int n    = row0 + lrow;
            dist_out[(size_t)b * nq + n] = minv[r] + qsq[lrow];
            idx_out [(size_t)b * nq + n] = mini[r];
        }
    }
}

extern "C" void kernel_launch(void* const* d_in, const int* in_sizes, int n_in,
                              void* d_out, int out_size, void* d_ws, size_t ws_size,
                              hipStream_t stream) {
    (void)in_sizes; (void)n_in; (void)out_size; (void)d_ws; (void)ws_size;
    const int B = 16, N = 4096, M = 4096;

    const float* in1 = (const float*)d_in[0];   // (B, N, 3)
    const float* in2 = (const float*)d_in[1];   // (B, M, 3)

    // Output order: dist1 (B*N f32), dist2 (B*M f32), idx1 (B*N i32), idx2 (B*M i32)
    float* out   = (float*)d_out;
    float* dist1 = out;
    float* dist2 = out + (size_t)B * N;
    int*   idx1  = (int*)(out + (size_t)B * N + (size_t)B * M);
    int*   idx2  = (int*)(out + (size_t)B * N + (size_t)B * M + (size_t)B * N);

    dim3 block(BLOCK_THREADS);
    dim3 grid1(N / ROWS_PER_BLOCK, B);
    dim3 grid2(M / ROWS_PER_BLOCK, B);

    // dist1/idx1: query = input1, ref = input2
    chamfer_nn_wmma<<<grid1, block, 0, stream>>>(in1, in2, N, M, dist1, idx1);
    // dist2/idx2: query = input2, ref = input1 (problem is symmetric)
    chamfer_nn_wmma<<<grid2, block, 0, stream>>>(in2, in1, M, N, dist2, idx2);
}